// VpSde_61787399520374
// MI455X (gfx1250) — compile-verified
//
#include <hip/hip_runtime.h>
#include <hip/hip_bf16.h>

// ---------------- constants matching the reference ----------------
#define GRID_N 10
#define T_STEPS 80
#define N_AGENTS 98304          // 256 * 64 * 6
#define WAVES_PER_BLOCK 8
#define NB_AGENT (N_AGENTS / WAVES_PER_BLOCK)   // 12288 blocks
#define FIELD_FLOATS (100 * 100)                // 100 fields x 100 cells

static __device__ __forceinline__ float dxc()   { return (float)(20.0 / 9.0); }
static __device__ __forceinline__ float denomc(){ return (float)(4.0 + 25.0 * (20.0/9.0) * (20.0/9.0)); }

typedef __attribute__((ext_vector_type(2))) float v2f;
typedef __attribute__((ext_vector_type(8))) float v8f;

// ---------------------------------------------------------------------------
// Kernel A: precompute the 100 possible Poisson fields with f32 WMMA.
// One wave per goal cell g. U is kept in the WMMA 16x16 f32 C/D layout:
//   vgpr i, lane l : M = i + 8*(l>=16), N = l & 15
// Jacobi: lap = S*U + U*S  where S[i][j] = 1 iff |i-j|==1 and i,j < 10.
// Each 16x16x16 product = 4 chained V_WMMA_F32_16X16X4_F32 accumulations.
// ---------------------------------------------------------------------------

// Gather element (M, N) of a D-layout matrix held in u[0..7] (per-lane).
static __device__ __forceinline__ float gatherD(const float u[8], int M, int N) {
  int src = (N & 15) | ((M >= 8) ? 16 : 0);
  int sel = M & 7;
  float r = __shfl(u[0], src, 32);
#pragma unroll
  for (int i = 1; i < 8; ++i) {
    float t = __shfl(u[i], src, 32);
    r = (sel == i) ? t : r;
  }
  return r;
}

static __device__ __forceinline__ float Sval(int i, int j) {
  int d = i - j; if (d < 0) d = -d;
  return (d == 1 && i < 10 && j < 10) ? 1.0f : 0.0f;
}

__global__ __launch_bounds__(32) void fields_kernel(float* __restrict__ F) {
  const int g    = blockIdx.x;          // goal cell 0..99
  const int gy   = g / 10, gx = g % 10;
  const int lane = threadIdx.x;         // wave32
  const int hi   = lane >> 4;           // 0: lanes 0-15, 1: lanes 16-31
  const int col  = lane & 15;

  float u[8];
#pragma unroll
  for (int i = 0; i < 8; ++i) {
    int M = i + 8 * hi, N = col;
    float v = (M < 10 && N < 10) ? 0.5f : 0.0f;
    if (M == gy && N == gx) v = 1.0f;
    u[i] = v;
  }

  for (int it = 0; it < 10; ++it) {
    v8f acc = {0.f, 0.f, 0.f, 0.f, 0.f, 0.f, 0.f, 0.f};

    // S * U : A = S (formulaic, A-layout), B = U (D->B layout via shuffles)
#pragma unroll
    for (int k = 0; k < 4; ++k) {
      v2f a, b;
      a.x = Sval(col, 4 * k + 0 + 2 * hi);
      a.y = Sval(col, 4 * k + 1 + 2 * hi);
      b.x = gatherD(u, 4 * k + 0 + 2 * hi, col);
      b.y = gatherD(u, 4 * k + 1 + 2 * hi, col);
      acc = __builtin_amdgcn_wmma_f32_16x16x4_f32(false, a, false, b,
                                                  (short)0, acc, false, false);
    }
    // + U * S : A = U (D->A layout via shuffles), B = S (formulaic)
#pragma unroll
    for (int k = 0; k < 4; ++k) {
      v2f a, b;
      a.x = gatherD(u, col, 4 * k + 0 + 2 * hi);
      a.y = gatherD(u, col, 4 * k + 1 + 2 * hi);
      b.x = Sval(4 * k + 0 + 2 * hi, col);
      b.y = Sval(4 * k + 1 + 2 * hi, col);
      acc = __builtin_amdgcn_wmma_f32_16x16x4_f32(false, a, false, b,
                                                  (short)0, acc, false, false);
    }

#pragma unroll
    for (int i = 0; i < 8; ++i) {
      int M = i + 8 * hi, N = col;
      float v = (M < 10 && N < 10) ? (acc[i] / denomc()) : 0.0f;
      if (M == gy && N == gx) v = 1.0f;   // boundary cell stays pinned
      u[i] = v;
    }
  }

#pragma unroll
  for (int i = 0; i < 8; ++i) {
    int M = i + 8 * hi, N = col;
    if (M < 10 && N < 10) F[g * 100 + M * 10 + N] = u[i];
  }
}

// ---------------------------------------------------------------------------
// Kernel B: one wave per agent. Stage the agent's field into LDS, coalesced
// trajectory sampling (lane l handles t = l, l+32, l+64), wave32 reduction,
// per-block partial sum.
// ---------------------------------------------------------------------------
__global__ __launch_bounds__(256) void agent_kernel(
    const float* __restrict__ pos,    // [N,80,2]
    const float* __restrict__ gw,     // [N,3]
    const float* __restrict__ head,   // [N,2]
    const float* __restrict__ F,      // [100,100]
    float* __restrict__ partials)     // [NB_AGENT]
{
  __shared__ float fld[WAVES_PER_BLOCK][100];
  __shared__ float wpart[WAVES_PER_BLOCK];

  const int lane = threadIdx.x & 31;
  const int wv   = threadIdx.x >> 5;
  const int n    = blockIdx.x * WAVES_PER_BLOCK + wv;

  const float* P = pos + (size_t)n * (2 * T_STEPS);
  __builtin_prefetch(P, 0, 1);   // global_prefetch_b8: pull trajectory line in

  // per-agent scalars (uniform across the wave; broadcast via cache)
  float hx = head[2 * n + 0], hy = head[2 * n + 1];
  float nn = fmaxf(sqrtf(hx * hx + hy * hy), 1e-12f);
  hx /= nn; hy /= nn;

  float p0x = P[0], p0y = P[1];
  float pTx = P[2 * (T_STEPS - 1)], pTy = P[2 * (T_STEPS - 1) + 1];
  float dpx = pTx - p0x, dpy = pTy - p0y;
  float ld  = dpx * (-hy) + dpy * hx;          // left drift
  float eL  = -ld + fmaxf(-ld, 0.0f);          // PENALTY = 1
  float eR  =  ld + fmaxf( ld, 0.0f);

  float gx = fminf(fmaxf(p0x + hx * 5.0f, -10.0f), 10.0f);
  float gy = fminf(fmaxf(p0y + hy * 5.0f, -10.0f), 10.0f);
  int gxi = (int)((gx + 10.0f) / dxc());
  int gyi = (int)((gy + 10.0f) / dxc());
  gxi = min(max(gxi, 0), 9);
  gyi = min(max(gyi, 0), 9);
  const int g = gyi * 10 + gxi;

  // stage this agent's 100-float field into LDS
  const float* Fg = F + g * 100;
  for (int i = lane; i < 100; i += 32) fld[wv][i] = Fg[i];
  __syncthreads();

  // bilinear sampling over the 80 trajectory points
  float s = 0.0f;
  for (int t = lane; t < T_STEPS; t += 32) {
    float px = P[2 * t + 0];
    float py = P[2 * t + 1];
    float ix = fminf(fmaxf((px + 10.0f) * 0.45f, 0.0f), 9.0f);
    float iy = fminf(fmaxf((py + 10.0f) * 0.45f, 0.0f), 9.0f);
    float x0 = floorf(ix), y0 = floorf(iy);
    float wx = ix - x0,   wy = iy - y0;
    int x0i = (int)x0, y0i = (int)y0;
    int x1i = min(x0i + 1, 9), y1i = min(y0i + 1, 9);
    float v00 = fld[wv][y0i * 10 + x0i];
    float v01 = fld[wv][y0i * 10 + x1i];
    float v10 = fld[wv][y1i * 10 + x0i];
    float v11 = fld[wv][y1i * 10 + x1i];
    s += (1.0f - wy) * ((1.0f - wx) * v00 + wx * v01)
       +         wy  * ((1.0f - wx) * v10 + wx * v11);
  }

  // wave32 reduction
#pragma unroll
  for (int o = 16; o > 0; o >>= 1) s += __shfl_down(s, o, 32);

  if (lane == 0) {
    float w0 = gw[3 * n + 0], w1 = gw[3 * n + 1], w2 = gw[3 * n + 2];
    float eLane = (w0 > 0.0001f) ? -s : 0.0f;
    wpart[wv] = w0 * eLane + w1 * eL + w2 * eR;
  }
  __syncthreads();

  if (threadIdx.x == 0) {
    float b = 0.0f;
#pragma unroll
    for (int i = 0; i < WAVES_PER_BLOCK; ++i) b += wpart[i];
    partials[blockIdx.x] = b;
  }
}

// ---------------------------------------------------------------------------
// Kernel C: deterministic reduction of the block partials.
// ---------------------------------------------------------------------------
__global__ __launch_bounds__(256) void reduce_kernel(
    const float* __restrict__ partials, float* __restrict__ out, int n)
{
  __shared__ float red[256];
  float s = 0.0f;
  for (int i = threadIdx.x; i < n; i += 256) s += partials[i];
  red[threadIdx.x] = s;
  __syncthreads();
  for (int off = 128; off > 0; off >>= 1) {
    if ((int)threadIdx.x < off) red[threadIdx.x] += red[threadIdx.x + off];
    __syncthreads();
  }
  if (threadIdx.x == 0) out[0] = red[0];
}

// ---------------------------------------------------------------------------
extern "C" void kernel_launch(void* const* d_in, const int* in_sizes, int n_in,
                              void* d_out, int out_size, void* d_ws, size_t ws_size,
                              hipStream_t stream) {
  (void)in_sizes; (void)n_in; (void)out_size; (void)ws_size;
  const float* pos  = (const float*)d_in[0];  // agent_positions_flat [N,80,2]
  const float* gw   = (const float*)d_in[1];  // guidance_weights     [B,A,I,3]
  const float* head = (const float*)d_in[2];  // initial_headings     [N,2]
  // d_in[3] map_points: unused by the reference
  float* F        = (float*)d_ws;             // 100x100 fields (40 KB)
  float* partials = F + FIELD_FLOATS;         // NB_AGENT block partials

  fields_kernel<<<100, 32, 0, stream>>>(F);
  agent_kernel<<<NB_AGENT, 256, 0, stream>>>(pos, gw, head, F, partials);
  reduce_kernel<<<1, 256, 0, stream>>>(partials, (float*)d_out, NB_AGENT);
}